// ParallelComplexModalFilter_481036337245
// MI455X (gfx1250) — compile-verified
//
#include <hip/hip_runtime.h>
#include <hip/hip_bf16.h>

typedef __attribute__((ext_vector_type(2))) float v2f;
typedef __attribute__((ext_vector_type(8))) float v8f;

#define SEQ_L 4096

// One workgroup per channel d. 256 threads = 8 wave32.
// l = 256*t + 16*ii + j ; tile (ii,j) is a 16x16 WMMA output.
// Contraction: K-dim = 64 = [32 real modes | 32 imag modes], done as 16x V_WMMA_F32_16X16X4_F32.
__global__ __launch_bounds__(256)
void s4d_vandermonde_wmma(const float* __restrict__ log_dt,
                          const float* __restrict__ C_ri,
                          const float* __restrict__ inv_A_real,
                          const float* __restrict__ A_imag,
                          float* __restrict__ out)
{
    __shared__ float sdtAr[32], sdtAi[32], sCr[32], sCi[32];
    __shared__ float Bs[64][16];      // B[k][j] : shared per channel
    __shared__ float As[8][16][64];   // per-wave A tile: A[m][k]

    const int tid = threadIdx.x;
    const int d   = blockIdx.x;

    // --- per-channel mode parameters ---
    if (tid < 32) {
        float dt = __expf(log_dt[d]);
        float Ar = -__expf(inv_A_real[d * 32 + tid]);
        float Ai = A_imag[d * 32 + tid];
        sdtAr[tid] = Ar * dt;
        sdtAi[tid] = Ai * dt;
        sCr[tid]   = C_ri[(d * 32 + tid) * 2 + 0];
        sCi[tid]   = C_ri[(d * 32 + tid) * 2 + 1];
    }
    __syncthreads();

    // --- B panel: Bs[n][j] = Re(exp(dtA[n]*j)), Bs[32+n][j] = -Im(exp(dtA[n]*j)) ---
    for (int p = tid; p < 512; p += 256) {
        int n = p >> 4, j = p & 15;
        float er  = __expf(sdtAr[n] * (float)j);
        float ang = sdtAi[n] * (float)j;
        Bs[n][j]      =  er * __cosf(ang);
        Bs[32 + n][j] = -er * __sinf(ang);
    }
    __syncthreads();

    const int wave  = tid >> 5;
    const int lane  = tid & 31;
    const int m     = lane & 15;   // tile row (A) / tile col (B,D)
    const int half  = lane >> 4;   // 0 or 1
    const int nbase = half << 4;   // this lane builds modes [nbase, nbase+16)
    const int kbase = half << 1;   // WMMA f32 16x16x4: lanes 16-31 hold K=2,3

    for (int t = wave; t < 16; t += 8) {
        // --- build A tile: A[m][n] = Re(C[n]*exp(dtA[n]*(256t+16m))), A[m][32+n] = Im(...) ---
        float li = (float)(256 * t + 16 * m);
        #pragma unroll
        for (int q = 0; q < 16; ++q) {
            int n = nbase + q;
            float er  = __expf(sdtAr[n] * li);
            float ang = sdtAi[n] * li;
            float Kr = er * __cosf(ang);
            float Ki = er * __sinf(ang);
            float cr = sCr[n], ci = sCi[n];
            As[wave][m][n]      = cr * Kr - ci * Ki;
            As[wave][m][32 + n] = cr * Ki + ci * Kr;
        }
        __syncthreads();

        // --- 16x64 @ 64x16 via 16 chained V_WMMA_F32_16X16X4_F32 ---
        v8f acc = {};
        #pragma unroll
        for (int s = 0; s < 16; ++s) {
            int k0 = 4 * s + kbase;
            v2f a, b;
            a.x = As[wave][m][k0];
            a.y = As[wave][m][k0 + 1];
            b.x = Bs[k0][m];
            b.y = Bs[k0 + 1][m];
            acc = __builtin_amdgcn_wmma_f32_16x16x4_f32(
                false, a, false, b, (short)0, acc, false, false);
        }

        // --- D layout: lane holds rows r+8*half, col m -> l = 256t + 16*(r+8*half) + m ---
        float* o = out + (size_t)d * SEQ_L + 256 * t;
        #pragma unroll
        for (int r = 0; r < 8; ++r) {
            o[16 * (r + 8 * half) + m] = 2.0f * acc[r];
        }
        __syncthreads();
    }
}

extern "C" void kernel_launch(void* const* d_in, const int* in_sizes, int n_in,
                              void* d_out, int out_size, void* d_ws, size_t ws_size,
                              hipStream_t stream) {
    (void)in_sizes; (void)n_in; (void)d_ws; (void)ws_size; (void)out_size;
    const float* log_dt     = (const float*)d_in[0];
    const float* C_ri       = (const float*)d_in[1];
    const float* inv_A_real = (const float*)d_in[2];
    const float* A_imag     = (const float*)d_in[3];
    float* out = (float*)d_out;

    dim3 grid(1024), block(256);
    hipLaunchKernelGGL(s4d_vandermonde_wmma, grid, block, 0, stream,
                       log_dt, C_ri, inv_A_real, A_imag, out);
}